// GCN_86268713107994
// MI455X (gfx1250) — compile-verified
//
#include <hip/hip_runtime.h>
#include <hip/hip_bf16.h>

#define N_NODES 50000
#define N_EDGES 800000
#define N_GRAPHS 500
#define HID 128
#define OUTD 10
#define N_LAYERS 4

typedef float v2f __attribute__((ext_vector_type(2)));
typedef float v8f __attribute__((ext_vector_type(8)));
typedef int   v4i __attribute__((ext_vector_type(4)));
typedef __attribute__((address_space(1))) v4i* gv4i_p;   // global int4*
typedef __attribute__((address_space(3))) v4i* lv4i_p;   // LDS int4*

#if defined(__gfx1250__) && __has_builtin(__builtin_amdgcn_global_load_async_to_lds_b128)
#define USE_ASYNC_LDS 1
#else
#define USE_ASYNC_LDS 0
#endif

__device__ __forceinline__ void atomAddF(float* p, float v) {
  unsafeAtomicAdd(p, v);   // native global_atomic_add_f32 on gfx1250
}

__device__ __forceinline__ void wait_async_zero() {
#if defined(__gfx1250__)
#if __has_builtin(__builtin_amdgcn_s_wait_asynccnt)
  __builtin_amdgcn_s_wait_asynccnt(0);
#else
  asm volatile("s_wait_asynccnt 0x0" ::: "memory");
#endif
#endif
}

// ---------------- utility kernels ----------------
__global__ void zero_f32(float* __restrict__ p, int n) {
  int i = blockIdx.x * blockDim.x + threadIdx.x;
  if (i < n) p[i] = 0.0f;
}

__global__ void fill_f32(float* __restrict__ p, float v, int n) {
  int i = blockIdx.x * blockDim.x + threadIdx.x;
  if (i < n) p[i] = v;
}

__global__ void edge_degree(const int* __restrict__ col, float* __restrict__ deg, int e) {
  int i = blockIdx.x * blockDim.x + threadIdx.x;
  if (i < e) atomAddF(&deg[col[i]], 1.0f);
}

__global__ void make_dinv(float* __restrict__ deg, int n) {
  int i = blockIdx.x * blockDim.x + threadIdx.x;
  if (i < n) {
    float d = deg[i];
    deg[i] = (d > 0.0f) ? rsqrtf(fmaxf(d, 1e-12f)) : 0.0f;
  }
}

// ---------------- WMMA GEMM: H = X @ W  (nrows x 128) @ (128 x 128) --------
// block = 256 threads (8 waves); block tile = 64 rows x 128 cols.
// Wave w owns 16 cols; register-blocks 4 row-subtiles (4 accumulators), so
// each B fragment is reused 4x and the 4 WMMAs per K-step are independent.
// Tiles staged to LDS via gfx1250 async-to-LDS b128 (ASYNCcnt) when available.
__global__ __launch_bounds__(256) void gemm_xw(const float* __restrict__ X,
                                               const float* __restrict__ W,
                                               float* __restrict__ H, int nrows) {
  extern __shared__ float smem[];
  float* Wl = smem;                 // 128*128
  float* Xl = smem + 128 * 128;     // 64 rows, padded stride 132 (conflict-free)
  const int tid = threadIdx.x;
  const int row0 = blockIdx.x * 64;

#if USE_ASYNC_LDS
  // W: 16384 floats = 4096 x b128; 16 per thread.
#pragma unroll
  for (int it = 0; it < 16; ++it) {
    int q = it * 256 + tid;                       // b128 index
    __builtin_amdgcn_global_load_async_to_lds_b128(
        (gv4i_p)(void*)(W + q * 4), (lv4i_p)(void*)(Wl + q * 4), 0, 0);
  }
  // X tile: 64 rows x 32 b128, padded LDS stride 132; 8 per thread.
#pragma unroll
  for (int it = 0; it < 8; ++it) {
    int q = it * 256 + tid;                       // 0..2047
    int r = q >> 5, c4 = q & 31;
    int gr = row0 + r;
    if (gr >= nrows) gr = nrows - 1;              // clamp tail (stores guarded)
    __builtin_amdgcn_global_load_async_to_lds_b128(
        (gv4i_p)(void*)(X + (size_t)gr * HID + c4 * 4),
        (lv4i_p)(void*)(Xl + r * 132 + c4 * 4), 0, 0);
  }
  wait_async_zero();
#else
  for (int i = tid; i < 128 * 128; i += 256) Wl[i] = W[i];
  for (int i = tid; i < 64 * 128; i += 256) {
    int r = i >> 7, c = i & 127;
    int gr = row0 + r;
    if (gr >= nrows) gr = nrows - 1;
    Xl[r * 132 + c] = X[(size_t)gr * HID + c];
  }
#endif
  __syncthreads();

  const int wave = tid >> 5;
  const int lane = tid & 31;
  const int col0 = wave * 16;
  const int lm = lane & 15;          // M (A) / N (B)
  const int lk = (lane >> 4) * 2;    // K sub-offset: lanes 0-15 -> K0,1; 16-31 -> K2,3

  v8f c0 = {}, c1 = {}, c2 = {}, c3 = {};
  for (int k = 0; k < HID; k += 4) {
    v2f b;
    b.x = Wl[(k + lk) * 128 + col0 + lm];
    b.y = Wl[(k + lk + 1) * 128 + col0 + lm];
    const int xoff = lm * 132 + k + lk;
    v2f a0, a1, a2, a3;
    a0.x = Xl[xoff];              a0.y = Xl[xoff + 1];
    a1.x = Xl[16 * 132 + xoff];   a1.y = Xl[16 * 132 + xoff + 1];
    a2.x = Xl[32 * 132 + xoff];   a2.y = Xl[32 * 132 + xoff + 1];
    a3.x = Xl[48 * 132 + xoff];   a3.y = Xl[48 * 132 + xoff + 1];
    c0 = __builtin_amdgcn_wmma_f32_16x16x4_f32(false, a0, false, b, (short)0, c0, false, false);
    c1 = __builtin_amdgcn_wmma_f32_16x16x4_f32(false, a1, false, b, (short)0, c1, false, false);
    c2 = __builtin_amdgcn_wmma_f32_16x16x4_f32(false, a2, false, b, (short)0, c2, false, false);
    c3 = __builtin_amdgcn_wmma_f32_16x16x4_f32(false, a3, false, b, (short)0, c3, false, false);
  }

  const int ocol = col0 + lm;
  const int rbase = row0 + (lane >> 4) * 8;
  v8f acc[4] = {c0, c1, c2, c3};
#pragma unroll
  for (int t = 0; t < 4; ++t) {
#pragma unroll
    for (int r = 0; r < 8; ++r) {
      int grow = rbase + t * 16 + r;
      if (grow < nrows) H[(size_t)grow * HID + ocol] = acc[t][r];
    }
  }
}

// ---------------- edge scatter: acc[col] += norm * h[row] -------------------
// one wave per edge per iteration; lane handles 4 consecutive floats (float4).
// Edge index / norm loads forced wave-uniform -> SMEM path, keeping VMEM free
// for the 512B gather + 512B of fp32 atomics per edge.
__global__ __launch_bounds__(256) void scatter_edges(
    const float* __restrict__ H, const int* __restrict__ row_idx,
    const int* __restrict__ col_idx, const float* __restrict__ dinv,
    float* __restrict__ acc, int n_edges, int n_nodes) {
  const int lane = threadIdx.x & 31;
  const int wid = (blockIdx.x * blockDim.x + threadIdx.x) >> 5;
  const int nwaves = (gridDim.x * blockDim.x) >> 5;
  const int total = n_edges + n_nodes;
  const float4* __restrict__ H4 = (const float4*)H;
  for (int e0 = wid; e0 < total; e0 += nwaves) {
    const int e = __builtin_amdgcn_readfirstlane(e0);   // wave-uniform
    int r, cl;
    if (e < n_edges) { r = row_idx[e]; cl = col_idx[e]; }
    else             { r = e - n_edges; cl = r; }       // self loop
    const float nrm = dinv[r] * dinv[cl];
    const float4 hv = H4[(size_t)r * 32 + lane];
    float* dst = acc + (size_t)cl * HID + lane * 4;
    atomAddF(dst + 0, nrm * hv.x);
    atomAddF(dst + 1, nrm * hv.y);
    atomAddF(dst + 2, nrm * hv.z);
    atomAddF(dst + 3, nrm * hv.w);
  }
}

// x = relu(acc + b[j]) for 128-wide rows
__global__ void finalize_layer(const float* __restrict__ acc,
                               const float* __restrict__ b,
                               float* __restrict__ x, int n) {
  int i = blockIdx.x * blockDim.x + threadIdx.x;
  if (i < n) x[i] = fmaxf(acc[i] + b[i & (HID - 1)], 0.0f);
}

// ---------------- pooling ----------------
__global__ void pool_nodes(const float* __restrict__ x, const int* __restrict__ batch,
                           float* __restrict__ pooled, int n) {
  int i = blockIdx.x * blockDim.x + threadIdx.x;
  if (i < n) {
    int node = i >> 7, j = i & 127;
    atomAddF(&pooled[(size_t)batch[node] * HID + j], x[i]);
  }
}

__global__ void pool_counts(const int* __restrict__ batch, float* __restrict__ counts, int n) {
  int i = blockIdx.x * blockDim.x + threadIdx.x;
  if (i < n) atomAddF(&counts[batch[i]], 1.0f);
}

// gmean = pooled / max(count, 1)
__global__ void pool_mean(const float* __restrict__ pooled, const float* __restrict__ counts,
                          float* __restrict__ gmean, int n) {
  int i = blockIdx.x * blockDim.x + threadIdx.x;
  if (i < n) gmean[i] = pooled[i] / fmaxf(counts[i >> 7], 1.0f);
}

// final 128 -> 10 projection (tiny; not WMMA-shaped)
__global__ void mlp2(const float* __restrict__ h1, const float* __restrict__ w,
                     const float* __restrict__ b, float* __restrict__ out) {
  int i = blockIdx.x * blockDim.x + threadIdx.x;
  if (i >= N_GRAPHS * OUTD) return;
  int g = i / OUTD, o = i % OUTD;
  float s = b[o];
  const float* hp = h1 + (size_t)g * HID;
#pragma unroll 4
  for (int k = 0; k < HID; ++k) s += hp[k] * w[k * OUTD + o];
  out[i] = s;
}

extern "C" void kernel_launch(void* const* d_in, const int* in_sizes, int n_in,
                              void* d_out, int out_size, void* d_ws, size_t ws_size,
                              hipStream_t stream) {
  const float* x0      = (const float*)d_in[0];
  const int*   erow    = (const int*)d_in[1];             // edge_index[0]
  const int*   ecol    = ((const int*)d_in[1]) + N_EDGES; // edge_index[1]
  const int*   batch   = (const int*)d_in[2];
  const float* conv_w  = (const float*)d_in[3];
  const float* conv_b  = (const float*)d_in[4];
  const float* lin1_w  = (const float*)d_in[5];
  const float* lin1_b  = (const float*)d_in[6];
  const float* lin2_w  = (const float*)d_in[7];
  const float* lin2_b  = (const float*)d_in[8];
  float* out = (float*)d_out;

  char* base = (char*)d_ws;
  const size_t NH = (size_t)N_NODES * HID;
  float* h      = (float*)(base);                      // N*H
  float* acc    = (float*)(base + NH * 4);             // N*H
  float* xbuf   = (float*)(base + NH * 8);             // N*H
  float* dinv   = (float*)(base + NH * 12);            // N (deg -> d^-1/2)
  float* pooled = (float*)(base + NH * 12 + 200000);   // G*H (+counts after)
  float* counts = pooled + (size_t)N_GRAPHS * HID;
  float* gmean  = counts + N_GRAPHS;                   // G*H
  float* h1raw  = gmean + (size_t)N_GRAPHS * HID;      // G*H
  float* h1     = h1raw + (size_t)N_GRAPHS * HID;      // G*H

  const int T = 256;
  const size_t gemm_lds = (128 * 128 + 64 * 132) * sizeof(float);  // ~97 KB

  // degrees (self-loop contributes 1 to every node) -> d^-1/2
  fill_f32<<<(N_NODES + T - 1) / T, T, 0, stream>>>(dinv, 1.0f, N_NODES);
  edge_degree<<<(N_EDGES + T - 1) / T, T, 0, stream>>>(ecol, dinv, N_EDGES);
  make_dinv<<<(N_NODES + T - 1) / T, T, 0, stream>>>(dinv, N_NODES);

  const int gemm_blocks = (N_NODES + 63) / 64;   // 782
  const float* xin = x0;
  for (int l = 0; l < N_LAYERS; ++l) {
    gemm_xw<<<gemm_blocks, T, gemm_lds, stream>>>(xin, conv_w + (size_t)l * HID * HID,
                                                  h, N_NODES);
    zero_f32<<<(int)((NH + T - 1) / T), T, 0, stream>>>(acc, (int)NH);
    scatter_edges<<<2048, T, 0, stream>>>(h, erow, ecol, dinv, acc, N_EDGES, N_NODES);
    finalize_layer<<<(int)((NH + T - 1) / T), T, 0, stream>>>(acc, conv_b + (size_t)l * HID,
                                                              xbuf, (int)NH);
    xin = xbuf;
  }

  // global mean pool
  zero_f32<<<(N_GRAPHS * HID + N_GRAPHS + T - 1) / T, T, 0, stream>>>(pooled,
                                                    N_GRAPHS * HID + N_GRAPHS);
  pool_nodes<<<(int)((NH + T - 1) / T), T, 0, stream>>>(xbuf, batch, pooled, (int)NH);
  pool_counts<<<(N_NODES + T - 1) / T, T, 0, stream>>>(batch, counts, N_NODES);
  pool_mean<<<(N_GRAPHS * HID + T - 1) / T, T, 0, stream>>>(pooled, counts, gmean,
                                                            N_GRAPHS * HID);

  // head: WMMA for the 500x128 @ 128x128 layer, scalar for 128->10
  gemm_xw<<<(N_GRAPHS + 63) / 64, T, gemm_lds, stream>>>(gmean, lin1_w, h1raw, N_GRAPHS);
  finalize_layer<<<(N_GRAPHS * HID + T - 1) / T, T, 0, stream>>>(h1raw, lin1_b, h1,
                                                                 N_GRAPHS * HID);
  mlp2<<<(N_GRAPHS * OUTD + T - 1) / T, T, 0, stream>>>(h1, lin2_w, lin2_b, out);
}